// GraphAutoencoder_43645457662003
// MI455X (gfx1250) — compile-verified
//
#include <hip/hip_runtime.h>

// GCN graph autoencoder for MI455X (gfx1250, wave32).
// - Encoder matmuls: V_WMMA_F32_16X16X4_F32, compile-time K/M, fully unrolled,
//   no per-load predication (tail rows handled by a scalar kernel, launched only
//   if N % 16 != 0 -- never for the reference shapes).
// - ReLU fused into the next layer's A-matrix loads.
// - Edge aggregation: hardware fp32 global atomics; whole working set is L2-resident.
// - Decoder collapsed to a single row (broadcast rows are identical), then b128 broadcast.

typedef __attribute__((ext_vector_type(2))) float v2f;
typedef __attribute__((ext_vector_type(8))) float v8f;

// ---------------- preprocessing ----------------

__global__ void deg_init_kernel(float* __restrict__ deg, int N) {
  int i = blockIdx.x * blockDim.x + threadIdx.x;
  if (i < N) deg[i] = 1.0f;  // self-loop contribution
}

__global__ void deg_scatter_kernel(const int* __restrict__ ei, float* __restrict__ deg, int E) {
  int e = blockIdx.x * blockDim.x + threadIdx.x;
  if (e < E) unsafeAtomicAdd(&deg[ei[E + e]], 1.0f);  // col = ei[E+e]
}

__global__ void dinv_kernel(float* __restrict__ deg, int N) {
  int i = blockIdx.x * blockDim.x + threadIdx.x;
  if (i < N) {
    float d = deg[i];
    deg[i] = (d > 0.f) ? rsqrtf(d) : 0.f;  // in-place: deg -> dinv
  }
}

__global__ void norm_kernel(const int* __restrict__ ei, const float* __restrict__ dinv,
                            float* __restrict__ norm, int E) {
  int e = blockIdx.x * blockDim.x + threadIdx.x;
  if (e < E) norm[e] = dinv[ei[e]] * dinv[ei[E + e]];
}

// ---------------- dense matmul: O[N,M] = H[N,K] @ W[K,M] via f32 WMMA ----------------
// One wave computes one 16x16 output tile over full row tiles (no bounds checks in
// the hot loop -> EXEC stays all-1s, loads merge into b64, K fully unrolled).

template <int K, int M, bool RELU_IN>
__global__ void gcn_matmul_wmma(const float* __restrict__ H, const float* __restrict__ W,
                                float* __restrict__ O, int nTiles) {
  int wave = (blockIdx.x * blockDim.x + threadIdx.x) >> 5;  // global wave id
  int lane = threadIdx.x & 31;
  constexpr int tilesM = M / 16;
  int tileN = wave / tilesM;           // row-tile (nodes)
  int tileM = wave - tileN * tilesM;   // col-tile (features)
  if (tileN >= nTiles) return;         // uniform per wave

  int m = lane & 15;
  int half = lane >> 4;                      // 0: K pair {0,1}; 1: K pair {2,3}
  const float* hp = H + (size_t)(tileN * 16 + m) * K + half * 2;  // A row
  const float* bp = W + half * 2 * M + (tileM * 16 + m);          // B column

  v8f c = {};
#pragma unroll
  for (int k = 0; k < K; k += 4) {
    v2f a = *(const v2f*)hp;           // contiguous 8B -> global_load_b64
    if (RELU_IN) {
      a.x = fmaxf(a.x, 0.f);
      a.y = fmaxf(a.y, 0.f);
    }
    v2f b;
    b.x = bp[0];
    b.y = bp[M];
    hp += 4;
    bp += 4 * M;
    c = __builtin_amdgcn_wmma_f32_16x16x4_f32(
        /*neg_a=*/false, a, /*neg_b=*/false, b,
        /*c_mod=*/(short)0, c, /*reuse_a=*/false, /*reuse_b=*/false);
  }

  // C/D layout: VGPR j -> row tileN*16 + half*8 + j, col tileM*16 + (lane&15)
  float* op = O + (size_t)(tileN * 16 + half * 8) * M + tileM * 16 + m;
#pragma unroll
  for (int j = 0; j < 8; ++j) op[j * M] = c[j];
}

// Scalar fallback for tail rows (only launched when N % 16 != 0).
__global__ void matmul_tail_kernel(const float* __restrict__ H, const float* __restrict__ W,
                                   float* __restrict__ O, int startRow, int N, int K, int M,
                                   int reluIn) {
  int t = blockIdx.x * blockDim.x + threadIdx.x;
  int row = startRow + t / M;
  int col = t - (t / M) * M;
  if (row >= N) return;
  float a = 0.f;
  for (int k = 0; k < K; ++k) {
    float h = H[(size_t)row * K + k];
    if (reluIn) h = fmaxf(h, 0.f);
    a += h * W[(size_t)k * M + col];
  }
  O[(size_t)row * M + col] = a;
}

// ---------------- GCN aggregation ----------------
// out[i] = dinv[i]^2 * hW[i] + bias   (self-loop + bias), then edge scatter-adds.

template <int LOGM>
__global__ void agg_init_kernel(const float* __restrict__ hW, const float* __restrict__ dinv,
                                const float* __restrict__ bias, float* __restrict__ out, int N) {
  constexpr int M = 1 << LOGM;
  int t = blockIdx.x * blockDim.x + threadIdx.x;
  int i = t >> LOGM;
  int f = t & (M - 1);
  if (i < N) {
    float dv = dinv[i];
    out[t] = dv * dv * hW[t] + bias[f];
  }
}

template <int LOGM>
__global__ void agg_scatter_kernel(const int* __restrict__ ei, const float* __restrict__ norm,
                                   const float* __restrict__ hW, float* __restrict__ out,
                                   int E) {
  constexpr int M = 1 << LOGM;
  int t = blockIdx.x * blockDim.x + threadIdx.x;
  int e = t >> LOGM;   // wave-uniform (LOGM >= 5, wave32)
  int f = t & (M - 1);
  if (e < E) {
    int r = ei[e];
    int c = ei[E + e];
    unsafeAtomicAdd(&out[((size_t)c << LOGM) + f], norm[e] * hW[((size_t)r << LOGM) + f]);
  }
}

// ---------------- deterministic mean pool over [N,32] ----------------

__global__ void pool_partial_kernel(const float* __restrict__ h, float* __restrict__ partials,
                                    int N) {
  __shared__ float s[256];
  int f = threadIdx.x & 31;  // feature
  int g = threadIdx.x >> 5;  // row group 0..7
  float acc = 0.f;
  for (int r = blockIdx.x * 8 + g; r < N; r += gridDim.x * 8) acc += h[(size_t)r * 32 + f];
  s[threadIdx.x] = acc;
  __syncthreads();
  if (g == 0) {
    float a = 0.f;
#pragma unroll
    for (int gg = 0; gg < 8; ++gg) a += s[gg * 32 + f];
    partials[blockIdx.x * 32 + f] = a;
  }
}

__global__ void pool_final_kernel(const float* __restrict__ partials, float* __restrict__ z,
                                  int nparts, float invN) {
  int f = threadIdx.x;  // 32 threads (one wave)
  float a = 0.f;
  for (int p = 0; p < nparts; ++p) a += partials[p * 32 + f];
  z[f] = a * invN;
}

// ---------------- decoder on a single row (broadcast rows identical) ----------------

__global__ void decoder_kernel(const float* __restrict__ z,
                               const float* __restrict__ W0, const float* __restrict__ b0,
                               const float* __restrict__ W1, const float* __restrict__ b1,
                               const float* __restrict__ W2, const float* __restrict__ b2,
                               float* __restrict__ dec_out, float* __restrict__ out_z) {
  __shared__ float zz[32], d0[64], d1[64];
  int t = threadIdx.x;  // 128 threads
  if (t < 32) {
    float v = z[t];
    zz[t] = v;
    out_z[t] = v;  // emit z to d_out[0..31]
  }
  __syncthreads();
  if (t < 64) {
    float a = b0[t];
    for (int k = 0; k < 32; ++k) a += zz[k] * W0[k * 64 + t];
    d0[t] = fmaxf(a, 0.f);
  }
  __syncthreads();
  if (t < 64) {
    float a = b1[t];
    for (int k = 0; k < 64; ++k) a += d0[k] * W1[k * 64 + t];
    d1[t] = fmaxf(a, 0.f);
  }
  __syncthreads();
  {
    float a = b2[t];
    for (int k = 0; k < 64; ++k) a += d1[k] * W2[k * 128 + t];
    dec_out[t] = a;
  }
}

__global__ void broadcast_kernel(const float4* __restrict__ dec4, float4* __restrict__ xr4,
                                 int total4) {
  int t = blockIdx.x * blockDim.x + threadIdx.x;
  if (t < total4) xr4[t] = dec4[t & 31];  // 128 floats = 32 float4 per row
}

// ---------------- host launch ----------------

extern "C" void kernel_launch(void* const* d_in, const int* in_sizes, int n_in,
                              void* d_out, int out_size, void* d_ws, size_t ws_size,
                              hipStream_t stream) {
  const float* x = (const float*)d_in[0];
  const int* ei = (const int*)d_in[1];
  const float* encW[3] = {(const float*)d_in[2], (const float*)d_in[3], (const float*)d_in[4]};
  const float* encB[3] = {(const float*)d_in[5], (const float*)d_in[6], (const float*)d_in[7]};
  const float* decW[3] = {(const float*)d_in[8], (const float*)d_in[9], (const float*)d_in[10]};
  const float* decB[3] = {(const float*)d_in[11], (const float*)d_in[12], (const float*)d_in[13]};

  const int N = in_sizes[0] / 128;
  const int E = in_sizes[1] / 2;

  float* ws = (float*)d_ws;
  size_t off = 0;
  float* bufA = ws + off; off += (size_t)N * 128;  // aggregated features (ping)
  float* bufB = ws + off; off += (size_t)N * 128;  // matmul output (pong)
  float* dinv = ws + off; off += (size_t)N;        // deg -> dinv in place
  float* norm = ws + off; off += (size_t)E;        // per-edge normalization
  float* partials = ws + off; off += 256 * 32;     // pool stage-1 partials
  float* zvec = ws + off; off += 32;               // pooled latent
  float* dec_out = ws + off; off += 128;           // decoded single row

  float* out = (float*)d_out;  // [0..31] = z, [32..] = xr flat

  // degree / dinv / per-edge norm (norm reused across all 3 layers)
  deg_init_kernel<<<(N + 255) / 256, 256, 0, stream>>>(dinv, N);
  deg_scatter_kernel<<<(E + 255) / 256, 256, 0, stream>>>(ei, dinv, E);
  dinv_kernel<<<(N + 255) / 256, 256, 0, stream>>>(dinv, N);
  norm_kernel<<<(E + 255) / 256, 256, 0, stream>>>(ei, dinv, norm, E);

  const int nTiles = N / 16;       // full 16-row tiles
  const int tailStart = nTiles * 16;
  const int tail = N - tailStart;  // 0 for the reference shapes

  // ---- layer 0: x[N,128] @ W0 -> [N,64], no input relu ----
  {
    constexpr int K = 128, M = 64, LOGM = 6;
    int waves = nTiles * (M / 16);
    gcn_matmul_wmma<K, M, false><<<(waves + 7) / 8, 256, 0, stream>>>(x, encW[0], bufB, nTiles);
    if (tail)
      matmul_tail_kernel<<<(tail * M + 255) / 256, 256, 0, stream>>>(x, encW[0], bufB, tailStart,
                                                                     N, K, M, 0);
    agg_init_kernel<LOGM><<<(N * M + 255) / 256, 256, 0, stream>>>(bufB, dinv, encB[0], bufA, N);
    agg_scatter_kernel<LOGM><<<((size_t)E * M + 255) / 256, 256, 0, stream>>>(ei, norm, bufB,
                                                                              bufA, E);
  }
  // ---- layer 1: relu(h)[N,64] @ W1 -> [N,64] (relu fused into A loads) ----
  {
    constexpr int K = 64, M = 64, LOGM = 6;
    int waves = nTiles * (M / 16);
    gcn_matmul_wmma<K, M, true><<<(waves + 7) / 8, 256, 0, stream>>>(bufA, encW[1], bufB, nTiles);
    if (tail)
      matmul_tail_kernel<<<(tail * M + 255) / 256, 256, 0, stream>>>(bufA, encW[1], bufB,
                                                                     tailStart, N, K, M, 1);
    agg_init_kernel<LOGM><<<(N * M + 255) / 256, 256, 0, stream>>>(bufB, dinv, encB[1], bufA, N);
    agg_scatter_kernel<LOGM><<<((size_t)E * M + 255) / 256, 256, 0, stream>>>(ei, norm, bufB,
                                                                              bufA, E);
  }
  // ---- layer 2: relu(h)[N,64] @ W2 -> [N,32] (relu fused into A loads) ----
  {
    constexpr int K = 64, M = 32, LOGM = 5;
    int waves = nTiles * (M / 16);
    gcn_matmul_wmma<K, M, true><<<(waves + 7) / 8, 256, 0, stream>>>(bufA, encW[2], bufB, nTiles);
    if (tail)
      matmul_tail_kernel<<<(tail * M + 255) / 256, 256, 0, stream>>>(bufA, encW[2], bufB,
                                                                     tailStart, N, K, M, 1);
    agg_init_kernel<LOGM><<<(N * M + 255) / 256, 256, 0, stream>>>(bufB, dinv, encB[2], bufA, N);
    agg_scatter_kernel<LOGM><<<((size_t)E * M + 255) / 256, 256, 0, stream>>>(ei, norm, bufB,
                                                                              bufA, E);
  }

  // global mean pool (deterministic two-stage reduction)
  pool_partial_kernel<<<256, 256, 0, stream>>>(bufA, partials, N);
  pool_final_kernel<<<1, 32, 0, stream>>>(partials, zvec, 256, 1.0f / (float)N);

  // decoder on one row; broadcast result to all N rows (b128 stores)
  decoder_kernel<<<1, 128, 0, stream>>>(zvec, decW[0], decB[0], decW[1], decB[1], decW[2],
                                        decB[2], dec_out, out);
  int total4 = N * 32;  // N*128 floats / 4
  broadcast_kernel<<<(total4 + 255) / 256, 256, 0, stream>>>((const float4*)dec_out,
                                                             (float4*)(out + 32), total4);
}